// GCN_20066087207116
// MI455X (gfx1250) — compile-verified
//
#include <hip/hip_runtime.h>

#define N_NODES 50000
#define N_EDGES 600000
#define F       128        // F_IN == F_OUT == 128

typedef __attribute__((ext_vector_type(2))) float v2f;
typedef __attribute__((ext_vector_type(8))) float v8f;

__device__ __forceinline__ void atomicAddF32(float* p, float v) {
    // relaxed agent-scope fp32 atomic -> want global_atomic_add_f32 (no CAS loop)
    __hip_atomic_fetch_add(p, v, __ATOMIC_RELAXED, __HIP_MEMORY_SCOPE_AGENT);
}

// ---- K4 (placed first in file so the disasm snippet shows the atomic lowering)
// edge scatter: one wave per edge, float4 per lane (32*4 = 128 feats) ----
__global__ void k_scatter(const int* __restrict__ row, const int* __restrict__ col,
                          const float* __restrict__ h, const float* __restrict__ dinv,
                          float* __restrict__ out) {
    int wave = (blockIdx.x * blockDim.x + threadIdx.x) >> 5;
    int lane = threadIdx.x & 31;
    if (wave >= N_EDGES) return;

    int r = row[wave];
    int c = col[wave];
    float norm = dinv[r] * dinv[c];

    const float4* __restrict__ hs = (const float4*)(h + (size_t)r * F);
    float* __restrict__ od = out + (size_t)c * F + lane * 4;

    float4 v = hs[lane];
    atomicAddF32(od + 0, v.x * norm);
    atomicAddF32(od + 1, v.y * norm);
    atomicAddF32(od + 2, v.z * norm);
    atomicAddF32(od + 3, v.w * norm);
}

// ---- K0: zero the output accumulator, init degree to 1.0 (self loop) ----
__global__ void k_init(float* __restrict__ out, float* __restrict__ deg) {
    int i = blockIdx.x * blockDim.x + threadIdx.x;
    if (i < N_NODES * F) out[i] = 0.0f;
    if (i < N_NODES)     deg[i] = 1.0f;
}

// ---- K1: in-degree accumulation over edge targets ----
__global__ void k_degree(const int* __restrict__ col, float* __restrict__ deg) {
    int e = blockIdx.x * blockDim.x + threadIdx.x;
    if (e < N_EDGES) atomicAddF32(&deg[col[e]], 1.0f);
}

// ---- K2: dinv = rsqrt(deg)  (deg >= 1 always, so the >0 guard is moot) ----
__global__ void k_dinv(float* __restrict__ deg_to_dinv) {
    int i = blockIdx.x * blockDim.x + threadIdx.x;
    if (i < N_NODES) deg_to_dinv[i] = rsqrtf(deg_to_dinv[i]);
}

// ---- K3: h = x @ W via V_WMMA_F32_16X16X4_F32, one wave per 16x16 tile ----
// Tiles: 3125 (M) x 8 (N) = 25000 waves, 8 waves/block, 3125 blocks: exact fit,
// so EXEC is all-1s for every WMMA (ISA requirement).
__global__ void k_gemm(const float* __restrict__ x, const float* __restrict__ W,
                       float* __restrict__ h) {
    int wave  = (blockIdx.x * blockDim.x + threadIdx.x) >> 5;
    int lane  = threadIdx.x & 31;
    int tileM = wave >> 3;      // 0..3124
    int tileN = wave & 7;       // 0..7

    int m  = (tileM << 4) + (lane & 15);   // A: row per lane
    int n  = (tileN << 4) + (lane & 15);   // B: col per lane
    int kb = (lane >> 4) << 1;             // lanes 0-15 -> K+{0,1}; 16-31 -> K+{2,3}

    const float* __restrict__ xrow = x + (size_t)m * F;

    v8f c = {};
    #pragma unroll
    for (int k0 = 0; k0 < F; k0 += 4) {
        v2f a, b;
        a.x = xrow[k0 + kb];
        a.y = xrow[k0 + kb + 1];
        b.x = W[(size_t)(k0 + kb)     * F + n];
        b.y = W[(size_t)(k0 + kb + 1) * F + n];
        c = __builtin_amdgcn_wmma_f32_16x16x4_f32(
                /*neg_a=*/false, a, /*neg_b=*/false, b,
                /*c_mod=*/(short)0, c, /*reuse_a=*/false, /*reuse_b=*/false);
    }

    // C/D layout: VGPR g -> M = g (lanes 0-15) or g+8 (lanes 16-31), N = lane%16
    int mbase = (tileM << 4) + ((lane >> 4) << 3);
    int ncol  = (tileN << 4) + (lane & 15);
    #pragma unroll
    for (int g = 0; g < 8; ++g)
        h[(size_t)(mbase + g) * F + ncol] = c[g];
}

// ---- K5: self-loop contribution + bias + ReLU (in-place on out) ----
__global__ void k_final(const float* __restrict__ h, const float* __restrict__ dinv,
                        const float* __restrict__ bias, float* __restrict__ out) {
    int idx = blockIdx.x * blockDim.x + threadIdx.x;
    if (idx < N_NODES * F) {
        int i = idx >> 7;       // node
        int f = idx & (F - 1);  // feature
        float d = dinv[i];
        float v = out[idx] + h[idx] * (d * d) + bias[f];
        out[idx] = fmaxf(v, 0.0f);
    }
}

extern "C" void kernel_launch(void* const* d_in, const int* in_sizes, int n_in,
                              void* d_out, int out_size, void* d_ws, size_t ws_size,
                              hipStream_t stream) {
    const float* x    = (const float*)d_in[0];
    const int*   ei   = (const int*)d_in[1];    // JAX default: int64 demoted to int32
    const float* W    = (const float*)d_in[2];
    const float* bias = (const float*)d_in[3];
    float*       out  = (float*)d_out;

    const int* row = ei;             // edge_index[0] : sources
    const int* col = ei + N_EDGES;   // edge_index[1] : targets

    float* dinv = (float*)d_ws;                 // N_NODES floats (deg -> dinv in place)
    float* h    = dinv + N_NODES;               // N_NODES*F floats (16B-aligned: 200000 % 16 == 0)

    // K0: zero out accumulator, deg = 1
    k_init<<<(N_NODES * F + 255) / 256, 256, 0, stream>>>(out, dinv);
    // K1: degree accumulation
    k_degree<<<(N_EDGES + 255) / 256, 256, 0, stream>>>(col, dinv);
    // K2: dinv = rsqrt(deg)
    k_dinv<<<(N_NODES + 255) / 256, 256, 0, stream>>>(dinv);
    // K3: h = x @ W  (25000 waves, 8 per block)
    k_gemm<<<3125, 256, 0, stream>>>(x, W, h);
    // K4: scatter-add messages (one wave per edge, 8 edges per block)
    k_scatter<<<(N_EDGES + 7) / 8, 256, 0, stream>>>(row, col, h, dinv, out);
    // K5: self loop + bias + relu
    k_final<<<(N_NODES * F + 255) / 256, 256, 0, stream>>>(h, dinv, bias, out);
}